// MultiHeadAttention_34016140984482
// MI455X (gfx1250) — compile-verified
//
#include <hip/hip_runtime.h>
#include <hip/hip_bf16.h>
#include <math.h>
#include <stdint.h>

// Problem constants (from the reference)
#define B_  4
#define S_  2048
#define D_  512
#define H_  8
#define DK_ 12
#define DV_ 32

typedef __attribute__((ext_vector_type(2))) float v2f;
typedef __attribute__((ext_vector_type(8))) float v8f;

// ---------------------------------------------------------------------------
// Generic GEMM + bias:  C[M x N] = A[M x K] * W[K x N] + bias[N]
// One wave per 16x16 output tile, K stepped by 4 via V_WMMA_F32_16X16X4_F32.
// A-layout (16x4, 32b): lane<16 -> row=lane, K={0,1}; lane>=16 -> row=lane-16, K={2,3}
// B-layout (4x16, 32b): lane<16 -> col=lane, K={0,1}; lane>=16 -> col=lane-16, K={2,3}
// C/D (16x16, 32b):     vgpr r, lane -> row = r + 8*(lane>>4), col = lane&15
// ---------------------------------------------------------------------------
__global__ void gemm_bias_wmma(const float* __restrict__ A,
                               const float* __restrict__ W,
                               const float* __restrict__ bias,
                               float* __restrict__ C,
                               int M, int N, int K) {
  const int lane = threadIdx.x & 31;
  const int l16  = lane & 15;
  const int half = lane >> 4;
  const int m0   = blockIdx.x * 16;
  const int n0   = blockIdx.y * 16;

  v8f acc = {};
  const float* arow = A + (size_t)(m0 + l16) * K + 2 * half;
  const float* wrow = W + (size_t)(2 * half) * N + n0 + l16;

  for (int k = 0; k < K; k += 4) {
    v2f a, b;
    a.x = arow[k];
    a.y = arow[k + 1];
    b.x = wrow[(size_t)k * N];
    b.y = wrow[(size_t)(k + 1) * N];
    acc = __builtin_amdgcn_wmma_f32_16x16x4_f32(false, a, false, b,
                                                (short)0, acc, false, false);
  }

  const float bn = bias[n0 + l16];
#pragma unroll
  for (int r = 0; r < 8; ++r) {
    C[(size_t)(m0 + r + 8 * half) * N + n0 + l16] = acc[r] + bn;
  }
}

// ---------------------------------------------------------------------------
// Fused attention: one workgroup (8 waves / 256 threads) handles a 16-query
// stripe of one (b,h): scores -> mask -> exp into a 16x2048 LDS stripe,
// exact row-sums, in-LDS normalization, then the normalized stripe is DMA'd
// to global p_attn with GLOBAL_STORE_ASYNC_FROM_LDS_B128 (ASYNCcnt) while the
// waves overlap the P*V WMMA phase.  Dynamic LDS ~140 KB (<= 320 KB/WGP).
// ---------------------------------------------------------------------------
#define EPITCH 2052   // 2048 + 4 pad -> LDS bank-conflict-free column reads

__global__ void attn_kernel(const float* __restrict__ q,    // [B,S,H*DK]
                            const float* __restrict__ kmat, // [B,S,H*DK]
                            const float* __restrict__ v,    // [B,S,H*DV]
                            const int*   __restrict__ mask, // [B,1,S]
                            float* __restrict__ p_out,      // [B,H,S,S]
                            float* __restrict__ x_out)      // [B,S,H*DV]
{
  extern __shared__ float smem[];
  float* s_e    = smem;               // 16 x EPITCH  (exp scores -> normalized p)
  float* s_red  = s_e + 16 * EPITCH;  // 8 waves x 256 (PV partials)
  float* s_psum = s_red + 2048;       // 8 waves x 16 rows
  float* s_inv  = s_psum + 128;       // 16

  const int tid  = threadIdx.x;
  const int lane = tid & 31;
  const int w    = tid >> 5;
  const int l16  = lane & 15;
  const int half = lane >> 4;

  const int qt = blockIdx.x & 127;          // S/16 = 128 query tiles
  const int h  = (blockIdx.x >> 7) & 7;
  const int b  = blockIdx.x >> 10;
  const int qbase = qt * 16;

  const float scale = 0.28867513459481288f; // 1/sqrt(12)

  // Q tile operands are invariant across key tiles: keep in registers.
  const float* qrow = q + ((size_t)(b * S_) + qbase + l16) * (H_ * DK_)
                        + h * DK_ + 2 * half;
  v2f qa[3];
#pragma unroll
  for (int s = 0; s < 3; ++s) { qa[s].x = qrow[s * 4]; qa[s].y = qrow[s * 4 + 1]; }

  float psum[8];
#pragma unroll
  for (int r = 0; r < 8; ++r) psum[r] = 0.f;

  const int* mrow = mask + b * S_;

  // ---- Phase 1: scores -> masked exp -> LDS stripe; per-row partial sums ----
  for (int t = 0; t < 16; ++t) {
    const int key = w * 256 + t * 16 + l16;
    const float* krow = kmat + ((size_t)(b * S_) + key) * (H_ * DK_)
                             + h * DK_ + 2 * half;
    v8f acc = {};
#pragma unroll
    for (int s = 0; s < 3; ++s) {
      v2f kb2; kb2.x = krow[s * 4]; kb2.y = krow[s * 4 + 1];
      acc = __builtin_amdgcn_wmma_f32_16x16x4_f32(false, qa[s], false, kb2,
                                                  (short)0, acc, false, false);
    }
    const bool keep = (mrow[key] != 0);
#pragma unroll
    for (int r = 0; r < 8; ++r) {
      const float e = keep ? __expf(acc[r] * scale) : 0.f; // masked: exp(-1e9)->0
      s_e[(r + 8 * half) * EPITCH + key] = e;
      psum[r] += e;
    }
  }

  // Reduce row partial sums over the 16 lanes of each half-wave.
#pragma unroll
  for (int r = 0; r < 8; ++r) {
    float x = psum[r];
    x += __shfl_xor(x, 1, 32);
    x += __shfl_xor(x, 2, 32);
    x += __shfl_xor(x, 4, 32);
    x += __shfl_xor(x, 8, 32);
    psum[r] = x;
  }
  if (l16 == 0) {
#pragma unroll
    for (int r = 0; r < 8; ++r) s_psum[w * 16 + 8 * half + r] = psum[r];
  }
  __syncthreads();
  if (tid < 16) {
    float tot = 0.f;
#pragma unroll
    for (int ww = 0; ww < 8; ++ww) tot += s_psum[ww * 16 + tid];
    s_inv[tid] = (tot > 0.f) ? (1.f / tot) : 0.f;
  }
  __syncthreads();

  // ---- Normalize the stripe in place (so P*V needs no epilogue scale and
  //      the async DMA below streams out final p_attn values). --------------
  for (int row = 0; row < 16; ++row) {
    const float inv = s_inv[row];
    float* p = s_e + row * EPITCH + tid * 8;
    float4 x0 = *(const float4*)(p);
    float4 x1 = *(const float4*)(p + 4);
    x0.x *= inv; x0.y *= inv; x0.z *= inv; x0.w *= inv;
    x1.x *= inv; x1.y *= inv; x1.z *= inv; x1.w *= inv;
    *(float4*)(p)     = x0;
    *(float4*)(p + 4) = x1;
  }
  __syncthreads();

  // ---- Async DMA of normalized stripe: LDS -> global p_attn ---------------
  // Per wave: 2 rows x 16 chunks of 512B (32 lanes x 16B per issue), tracked
  // by ASYNCcnt and overlapped with the P*V WMMA phase below.
  {
    const size_t pbase = (((size_t)b * H_ + h) * S_ + qbase) * (size_t)S_;
    const uint32_t lbase = (uint32_t)(uintptr_t)s_e;  // LDS byte offset (low dword)
#pragma unroll 4
    for (int i = 0; i < 32; ++i) {
      const int r = 2 * w + (i >> 4);   // this wave's rows
      const int c = i & 15;             // 512B chunk within the row
      const float*  gp = p_out + pbase + (size_t)r * S_ + c * 128 + lane * 4;
      const uint32_t lp = lbase + (uint32_t)(r * (EPITCH * 4) + c * 512 + lane * 16);
      asm volatile("global_store_async_from_lds_b128 %0, %1, off"
                   :: "v"(gp), "v"(lp)
                   : "memory");
    }
  }

  // ---- Phase 2 (overlapped with the async store stream): x = P * V --------
  // Waves 0..3 -> DV cols 0..15 (K-split 4x512); waves 4..7 -> cols 16..31.
  const int ct = w >> 2;
  const int kw = w & 3;
  v8f xacc = {};
  const float* vcol = v + ((size_t)b * S_) * (H_ * DV_) + h * DV_ + ct * 16 + l16;
  for (int kb = kw * 512; kb < kw * 512 + 512; kb += 4) {
    if ((kb & 63) == 0)  // stride the V stream ahead (global_prefetch_b8)
      __builtin_prefetch(vcol + (size_t)(kb + 128) * (H_ * DV_), 0, 1);
    v2f a, bb;
    const float* e = s_e + l16 * EPITCH + kb + 2 * half;
    a.x = e[0];
    a.y = e[1];
    const size_t kkey = (size_t)kb + 2 * half;
    bb.x = vcol[kkey * (H_ * DV_)];
    bb.y = vcol[(kkey + 1) * (H_ * DV_)];
    xacc = __builtin_amdgcn_wmma_f32_16x16x4_f32(false, a, false, bb,
                                                 (short)0, xacc, false, false);
  }
#pragma unroll
  for (int r = 0; r < 8; ++r) s_red[w * 256 + r * 32 + lane] = xacc[r];
  __syncthreads();

  // Combine K-split partials and write x tile (16 x 32); stripe is already
  // normalized, so no 1/rowsum here.
  for (int idx = tid; idx < 512; idx += 256) {
    const int M  = idx >> 5;
    const int Nc = idx & 31;
    const int ctile = Nc >> 4;
    const int N  = Nc & 15;
    const int base = (M & 7) * 32 + (M >> 3) * 16 + N;
    float sum = 0.f;
#pragma unroll
    for (int ww = 0; ww < 4; ++ww) sum += s_red[(ctile * 4 + ww) * 256 + base];
    x_out[((size_t)(b * S_) + qbase + M) * (H_ * DV_) + h * DV_ + Nc] = sum;
  }

  // Drain the LDS->global DMA before the workgroup retires (S_ENDPGM would
  // also wait, but keep the counter discipline explicit).
  asm volatile("s_wait_asynccnt 0x0" ::: "memory");
}

// ---------------------------------------------------------------------------
extern "C" void kernel_launch(void* const* d_in, const int* in_sizes, int n_in,
                              void* d_out, int out_size, void* d_ws, size_t ws_size,
                              hipStream_t stream) {
  const float* query = (const float*)d_in[0];
  const float* key   = (const float*)d_in[1];
  const float* value = (const float*)d_in[2];
  const int*   mask  = (const int*)d_in[3];
  const float* Wq    = (const float*)d_in[4];
  const float* bq    = (const float*)d_in[5];
  const float* Wk    = (const float*)d_in[6];
  const float* bk    = (const float*)d_in[7];
  const float* Wv    = (const float*)d_in[8];
  const float* bv    = (const float*)d_in[9];
  const float* Wx    = (const float*)d_in[10];
  const float* bx    = (const float*)d_in[11];

  float* out_x = (float*)d_out;                      // [B,S,D]
  float* out_p = out_x + (size_t)B_ * S_ * D_;       // [B,H,S,S]

  float* ws   = (float*)d_ws;
  float* q_ws = ws;                                  // [B,S,H*DK]
  float* k_ws = q_ws + (size_t)B_ * S_ * H_ * DK_;   // [B,S,H*DK]
  float* v_ws = k_ws + (size_t)B_ * S_ * H_ * DK_;   // [B,S,H*DV]
  float* x_ws = v_ws + (size_t)B_ * S_ * H_ * DV_;   // [B,S,H*DV]

  const int M = B_ * S_;  // 8192 rows
  dim3 wv(32);

  // Projections
  gemm_bias_wmma<<<dim3(M / 16, (H_ * DK_) / 16), wv, 0, stream>>>(
      query, Wq, bq, q_ws, M, H_ * DK_, D_);
  gemm_bias_wmma<<<dim3(M / 16, (H_ * DK_) / 16), wv, 0, stream>>>(
      key, Wk, bk, k_ws, M, H_ * DK_, D_);
  gemm_bias_wmma<<<dim3(M / 16, (H_ * DV_) / 16), wv, 0, stream>>>(
      value, Wv, bv, v_ws, M, H_ * DV_, D_);

  // Fused attention (needs ~140 KB dynamic LDS; CDNA5 WGP has 320 KB)
  const size_t smem_bytes = (size_t)(16 * EPITCH + 2048 + 128 + 16) * sizeof(float);
  hipFuncSetAttribute((const void*)attn_kernel,
                      hipFuncAttributeMaxDynamicSharedMemorySize,
                      (int)smem_bytes);
  attn_kernel<<<dim3(B_ * H_ * (S_ / 16)), dim3(256), smem_bytes, stream>>>(
      q_ws, k_ws, v_ws, mask, out_p, x_ws);

  // Output projection
  gemm_bias_wmma<<<dim3(M / 16, D_ / 16), wv, 0, stream>>>(
      x_ws, Wx, bx, out_x, M, D_, H_ * DV_);
}